// KeypointBlock_77549929496882
// MI455X (gfx1250) — compile-verified
//
#include <hip/hip_runtime.h>
#include <math.h>

typedef __attribute__((ext_vector_type(2))) float v2f;
typedef __attribute__((ext_vector_type(8))) float v8f;

#define NB   4
#define HH   512
#define WW   640
#define CD   128
#define PHW  16
#define PPIX 256
#define LX   (WW / PHW)   // 40
#define LY   (HH / PHW)   // 32
#define LL   (LX * LY)    // 1280
#define ASTR 17           // LDS att row stride (conflict-free stores)

// d_out layout (flat, reference return order):
//  [0]                keypoints_2D    [N, L, 2]   = 10240
//  [10240]            keypoint_coords [N, 3, L]   = 15360
//  [25600]            keypoint_descs  [N, 128, L] = 655360
//  [680960]           keypoint_weights[N, 1, L]   = 5120
#define OFF_COORD 10240
#define OFF_DESC  25600
#define OFF_W     680960

__device__ inline float waveSum(float v) {
#pragma unroll
    for (int o = 16; o > 0; o >>= 1) v += __shfl_xor(v, o, 32);
    return v;
}
__device__ inline float waveMax(float v) {
#pragma unroll
    for (int o = 16; o > 0; o >>= 1) {
        float t = __shfl_xor(v, o, 32);
        v = v > t ? v : t;
    }
    return v;
}

// One workgroup (256 threads = 8 waves) handles 16 consecutive patches.
// Thread t is pixel t of the current patch (r = t/16, c = t%16).
__global__ __launch_bounds__(256)
void patch_softmax_kernel(const float* __restrict__ geom,
                          const float* __restrict__ det,
                          float* __restrict__ out2d,
                          float* __restrict__ outCoords)
{
    __shared__ float attBuf[PPIX * ASTR];      // att[pixel][patch], 17.4 KB
    __shared__ float pmax[8], psum[8], pg0[8], pg1[8], pg2[8];

    const int tid  = threadIdx.x;
    const int lane = tid & 31, wid = tid >> 5;
    const int n    = blockIdx.x / (LL / 16);
    const int grp  = blockIdx.x % (LL / 16);
    const int r = tid >> 4, c = tid & 15;

    for (int j = 0; j < 16; ++j) {
        const int l  = grp * 16 + j;
        const int py = l / LX, px = l % LX;
        const int y  = py * PHW + r, x = px * PHW + c;

        const float gx = geom[((long)(n * 3 + 0) * HH + y) * WW + x];
        const float gy = geom[((long)(n * 3 + 1) * HH + y) * WW + x];
        const float gz = geom[((long)(n * 3 + 2) * HH + y) * WW + x];
        float d = det[((long)n * HH + y) * WW + x];
        if (gx * gx + gy * gy + gz * gz == 0.0f) d = -20.0f;   // no_gap mask

        __syncthreads();                       // protect partial buffers
        float wm = waveMax(d);
        if (lane == 0) pmax[wid] = wm;
        __syncthreads();
        float m = pmax[0];
#pragma unroll
        for (int i = 1; i < 8; ++i) m = fmaxf(m, pmax[i]);

        float e  = expf(d - m);
        float ws = waveSum(e);
        if (lane == 0) psum[wid] = ws;
        __syncthreads();
        float s = 0.f;
#pragma unroll
        for (int i = 0; i < 8; ++i) s += psum[i];

        const float a = e / s;
        attBuf[tid * ASTR + j] = a;            // B-matrix column j, K-dim = pixel

        float s0 = waveSum(gx * a), s1 = waveSum(gy * a), s2 = waveSum(gz * a);
        if (lane == 0) { pg0[wid] = s0; pg1[wid] = s1; pg2[wid] = s2; }
        __syncthreads();
        if (tid == 0) {
            float t0 = 0.f, t1 = 0.f, t2 = 0.f;
#pragma unroll
            for (int i = 0; i < 8; ++i) { t0 += pg0[i]; t1 += pg1[i]; t2 += pg2[i]; }
            outCoords[(n * 3 + 0) * LL + l] = t0;
            outCoords[(n * 3 + 1) * LL + l] = t1;
            outCoords[(n * 3 + 2) * LL + l] = t2;
        }
    }
    __syncthreads();

    // Wave 0: D[2x16] = Coef[2x256] x Att[256x16] via v_wmma_f32_16x16x4_f32.
    // A row 0 = du(k)=k&15, row 1 = dv(k)=k>>4, rows 2..15 = 0 (shared coeffs).
    if (tid < 32) {
        v8f acc = {};
        const int m16   = lane & 15;           // A: M row / B: N col (patch)
        const int khalf = (lane >> 4) << 1;    // lanes 16-31 hold K+2, K+3
        for (int k0 = 0; k0 < PPIX; k0 += 4) {
            const int ka = k0 + khalf, kb = ka + 1;
            v2f A, B;
            A.x = (m16 == 0) ? (float)(ka & 15) : (m16 == 1) ? (float)(ka >> 4) : 0.f;
            A.y = (m16 == 0) ? (float)(kb & 15) : (m16 == 1) ? (float)(kb >> 4) : 0.f;
            B.x = attBuf[ka * ASTR + m16];
            B.y = attBuf[kb * ASTR + m16];
            acc = __builtin_amdgcn_wmma_f32_16x16x4_f32(
                false, A, false, B, (short)0, acc, false, false);
        }
        if (lane < 16) {                       // D rows 0,1 live in acc[0],acc[1], lanes 0-15
            const int l  = grp * 16 + lane;
            const int py = l / LX, px = l % LX;
            const float eu = (float)(px * PHW) + acc[0];  // base_u * sum(att)=1
            const float ev = (float)(py * PHW) + acc[1];
            out2d[((long)n * LL + l) * 2 + 0] = eu;
            out2d[((long)n * LL + l) * 2 + 1] = ev;
        }
    }
}

__device__ inline float fetch_zp(const float* __restrict__ p, int xi, int yi) {
    const bool valid = ((unsigned)xi < (unsigned)WW) && ((unsigned)yi < (unsigned)HH);
    const int xc = xi < 0 ? 0 : (xi > WW - 1 ? WW - 1 : xi);
    const int yc = yi < 0 ? 0 : (yi > HH - 1 ? HH - 1 : yi);
    const float v = p[(long)yc * WW + xc];
    return valid ? v : 0.f;
}

// Block = (n, channel) [channel 128 == weight plane]; threads stride over L
// so the [N,C,L] output writes are coalesced.
__global__ __launch_bounds__(256)
void sample_kernel(const float* __restrict__ desc,
                   const float* __restrict__ wsc,
                   const float* __restrict__ out2d,
                   float* __restrict__ outDesc,
                   float* __restrict__ outW)
{
    const int n  = blockIdx.x / (CD + 1);
    const int cc = blockIdx.x % (CD + 1);
    const float* plane = (cc < CD) ? (desc + (long)(n * CD + cc) * HH * WW)
                                   : (wsc + (long)n * HH * WW);
    float* dst = (cc < CD) ? (outDesc + (long)(n * CD + cc) * LL)
                           : (outW + (long)n * LL);

    for (int l = threadIdx.x; l < LL; l += 256) {
        const float eu = out2d[((long)n * LL + l) * 2 + 0];
        const float ev = out2d[((long)n * LL + l) * 2 + 1];
        // grid_sample align_corners=False with u_norm = 2u/(W-1)-1  =>  x = u*W/(W-1) - 0.5
        const float xf = eu * ((float)WW / (float)(WW - 1)) - 0.5f;
        const float yf = ev * ((float)HH / (float)(HH - 1)) - 0.5f;
        const float x0f = floorf(xf), y0f = floorf(yf);
        const int x0 = (int)x0f, y0 = (int)y0f, x1 = x0 + 1, y1 = y0 + 1;
        const float wx1 = xf - x0f, wx0 = 1.f - wx1;
        const float wy1 = yf - y0f, wy0 = 1.f - wy1;

        const float v00 = fetch_zp(plane, x0, y0);
        const float v10 = fetch_zp(plane, x1, y0);
        const float v01 = fetch_zp(plane, x0, y1);
        const float v11 = fetch_zp(plane, x1, y1);

        dst[l] = v00 * (wx0 * wy0) + v10 * (wx1 * wy0)
               + v01 * (wx0 * wy1) + v11 * (wx1 * wy1);
    }
}

extern "C" void kernel_launch(void* const* d_in, const int* in_sizes, int n_in,
                              void* d_out, int out_size, void* d_ws, size_t ws_size,
                              hipStream_t stream) {
    (void)in_sizes; (void)n_in; (void)out_size; (void)d_ws; (void)ws_size;
    const float* geom = (const float*)d_in[0];   // [4,3,512,640]
    const float* desc = (const float*)d_in[1];   // [4,128,512,640]
    const float* det  = (const float*)d_in[2];   // [4,1,512,640]
    const float* wsc  = (const float*)d_in[3];   // [4,1,512,640]
    float* out = (float*)d_out;

    float* out2d     = out;             // [4,1280,2]
    float* outCoords = out + OFF_COORD; // [4,3,1280]
    float* outDescs  = out + OFF_DESC;  // [4,128,1280]
    float* outW      = out + OFF_W;     // [4,1,1280]

    // Phase 1: softmax attention, WMMA coordinate expectation, geometry sums.
    patch_softmax_kernel<<<NB * (LL / 16), 256, 0, stream>>>(geom, det, out2d, outCoords);
    // Phase 2: bilinear descriptor/weight sampling at expected keypoints
    // (reads out2d written by phase 1; same stream => ordered).
    sample_kernel<<<NB * (CD + 1), 256, 0, stream>>>(desc, wsc, out2d, outDescs, outW);
}